// TensorProduct_33036888441470
// MI455X (gfx1250) — compile-verified
//
#include <hip/hip_runtime.h>
#include <hip/hip_bf16.h>
#include <math.h>

#define NINSTR 17
#define MULW   128
#define X1W    1152
#define X2W    16
#define WW     (NINSTR * MULW)   // 2176
#define OUTW   9088
#define W3JSTR 245               // padded 5*7*7 per instruction

typedef float v2f __attribute__((ext_vector_type(2)));
typedef float v4f __attribute__((ext_vector_type(4)));
typedef float v8f __attribute__((ext_vector_type(8)));

// ---------------- on-device Wigner-3j (real basis) table -------------------
__device__ float g_w3j[NINSTR * W3JSTR];

__constant__ int C_I1[NINSTR] = {0,1,2,0,1,1,2,2,0,1,1,2,2,0,1,2,2};
__constant__ int C_I2[NINSTR] = {0,1,2,1,0,2,1,3,2,1,3,0,2,3,2,1,3};
__constant__ int C_LO[NINSTR] = {0,0,0,1,1,1,1,1,2,2,2,2,2,3,3,3,3};

struct cdbl { double x, y; };
__device__ inline cdbl cmul(cdbl a, cdbl b) { return {a.x*b.x - a.y*b.y, a.x*b.y + a.y*b.x}; }
__device__ inline cdbl cconj(cdbl a)        { return {a.x, -a.y}; }

__device__ double dfact(int n) { double r = 1.0; for (int i = 2; i <= n; ++i) r *= (double)i; return r; }

__device__ double su2_cg(int j1, int j2, int j3, int m1, int m2, int m3) {
  if (m3 != m1 + m2) return 0.0;
  int vmin = -j1 + j2 + m3; if (-j1 + m1 > vmin) vmin = -j1 + m1; if (0 > vmin) vmin = 0;
  int vmax = j2 + j3 + m1;  if (j3 - j1 + j2 < vmax) vmax = j3 - j1 + j2; if (j3 + m3 < vmax) vmax = j3 + m3;
  double C = sqrt((2.0*j3 + 1.0) *
      dfact(j3+j1-j2) * dfact(j3-j1+j2) * dfact(j1+j2-j3) * dfact(j3+m3) * dfact(j3-m3) /
      (dfact(j1+j2+j3+1) * dfact(j1-m1) * dfact(j1+m1) * dfact(j2-m2) * dfact(j2+m2)));
  double S = 0.0;
  for (int v = vmin; v <= vmax; ++v) {
    double sgn = ((v + j2 + m2) & 1) ? -1.0 : 1.0;
    S += sgn * dfact(j2+j3+m1-v) * dfact(j1-m1+v) /
         (dfact(v) * dfact(j3-j1+j2-v) * dfact(j3+m3-v) * dfact(v+j1-j2-m3));
  }
  return C * S;
}

__device__ void build_q(int l, cdbl* q) {
  int d = 2*l + 1;
  for (int i = 0; i < d*d; ++i) q[i] = {0.0, 0.0};
  double s = 1.0 / sqrt(2.0);
  for (int m = -l; m < 0; ++m) {
    q[(l+m)*d + (l-m)] = { s, 0.0};
    q[(l+m)*d + (l+m)] = {0.0, -s};
  }
  q[l*d + l] = {1.0, 0.0};
  for (int m = 1; m <= l; ++m) {
    double sg = (m & 1) ? -1.0 : 1.0;
    q[(l+m)*d + (l+m)] = {sg*s, 0.0};
    q[(l+m)*d + (l-m)] = {0.0, sg*s};
  }
  cdbl ph;
  switch (l & 3) {                       // (-i)^l
    case 0: ph = { 1.0,  0.0}; break;
    case 1: ph = { 0.0, -1.0}; break;
    case 2: ph = {-1.0,  0.0}; break;
    default: ph = { 0.0,  1.0}; break;
  }
  for (int i = 0; i < d*d; ++i) q[i] = cmul(ph, q[i]);
}

__global__ void w3j_setup_kernel() {
  int k = threadIdx.x;
  if (k >= NINSTR) return;
  int l1 = C_I1[k], l2 = C_I2[k], l3 = C_LO[k];
  int d1 = 2*l1+1, d2 = 2*l2+1, d3 = 2*l3+1;

  double Ct[5*7*7];
  for (int i = 0; i < d1*d2*d3; ++i) Ct[i] = 0.0;
  for (int m1 = -l1; m1 <= l1; ++m1)
    for (int m2 = -l2; m2 <= l2; ++m2) {
      int m3 = m1 + m2;
      if (m3 >= -l3 && m3 <= l3)
        Ct[((l1+m1)*d2 + (l2+m2))*d3 + (l3+m3)] = su2_cg(l1, l2, l3, m1, m2, m3);
    }

  cdbl q1[49], q2[49], q3[49];
  build_q(l1, q1); build_q(l2, q2); build_q(l3, q3);

  double Cr[5*7*7];
  double nrm = 0.0;
  for (int a = 0; a < d1; ++a)
    for (int b = 0; b < d2; ++b)
      for (int c = 0; c < d3; ++c) {
        cdbl acc = {0.0, 0.0};
        for (int i = 0; i < d1; ++i)
          for (int kq = 0; kq < d2; ++kq)
            for (int n = 0; n < d3; ++n) {
              double coef = Ct[(i*d2 + kq)*d3 + n];
              if (coef == 0.0) continue;
              cdbl tt = cmul(q1[i*d1 + a], q2[kq*d2 + b]);
              tt = cmul(tt, cconj(q3[n*d3 + c]));
              acc.x += coef * tt.x; acc.y += coef * tt.y;
            }
        Cr[(a*d2 + b)*d3 + c] = acc.x;
        nrm += acc.x * acc.x;
      }
  double scale = sqrt((double)d3) / sqrt(nrm);

  float* dst = g_w3j + k * W3JSTR;
  for (int i = 0; i < W3JSTR; ++i) dst[i] = 0.f;
  for (int a = 0; a < d1; ++a)
    for (int b = 0; b < d2; ++b)
      for (int c = 0; c < d3; ++c)
        dst[a*49 + b*7 + c] = (float)(Cr[(a*d2 + b)*d3 + c] * scale);
}

// ---------------- main WMMA tensor-product kernel --------------------------
__constant__ unsigned char COLK[3][32] = {
  {0,3,3,3,8,8,8,8,8,13,13,13,13,13,13,13, 0,0,0,0,0,0,0,0,0,0,0,0,0,0,0,0},
  {1,4,4,4,5,5,5,9,9,9,9,9,10,10,10,10,10,14,14,14,14,14,14,14, 0,0,0,0,0,0,0,0},
  {2,6,6,6,7,7,7,11,11,11,11,11,12,12,12,12,12,15,15,15,15,15,15,15,16,16,16,16,16,16,16, 0}
};
__constant__ unsigned char COLKK[3][32] = {
  {0,0,1,2,0,1,2,3,4,0,1,2,3,4,5,6, 0,0,0,0,0,0,0,0,0,0,0,0,0,0,0,0},
  {0,0,1,2,0,1,2,0,1,2,3,4,0,1,2,3,4,0,1,2,3,4,5,6, 0,0,0,0,0,0,0,0},
  {0,0,1,2,0,1,2,0,1,2,3,4,0,1,2,3,4,0,1,2,3,4,5,6,0,1,2,3,4,5,6, 0}
};
__constant__ int K_NK[NINSTR]    = {1,1,1,3,3,3,3,3,5,5,5,5,5,7,7,7,7};
__constant__ int K_OUT[NINSTR]   = {0,128,256,384,768,1152,1536,1920,2304,2944,3584,4224,4864,5504,6400,7296,8192};
__constant__ int K_I2OFF[NINSTR] = {0,1,4,1,0,4,1,9,4,1,9,0,4,9,4,1,9};
__constant__ int K_NJ[NINSTR]    = {1,3,5,3,1,5,3,7,5,3,7,1,5,7,5,3,7};

// Low 32 bits of a generic pointer to LDS == wave-relative LDS byte address
// (FLAT aperture truncation rule, ISA sec.10.2).
__device__ __forceinline__ unsigned lds_off(const void* p) {
  return (unsigned)(unsigned long long)p;
}

__global__ __launch_bounds__(256) void tp_wmma_kernel(
    const float* __restrict__ x1, const float* __restrict__ x2,
    const float* __restrict__ wgt, float* __restrict__ out, int n) {
  __shared__ __align__(16) float s_x1[X1W];    //  4608 B, staged x1 row
  __shared__ __align__(16) float s_w[WW];      //  8704 B, staged weight row
  __shared__ __align__(16) float s_out[OUTW];  // 36352 B, staged output row
  __shared__ float s_xb[16];
  __shared__ float s_B[243];                   // [g0:1x16][g1:3x24][g2:5x31]

  int e = blockIdx.x;
  if (e >= n) return;
  int t = threadIdx.x;

  const float* xa   = x1  + (size_t)e * X1W;
  const float* wrow = wgt + (size_t)e * WW;
  float*       orow = out + (size_t)e * OUTW;

  const unsigned sx1b = lds_off(s_x1);
  const unsigned swb  = lds_off(s_w);
  const unsigned sob  = lds_off(s_out);

  // --- async-stage x1 row and weight row into LDS (coalesced B128) ---------
  for (int i = t; i < X1W/4; i += 256)
    asm volatile("global_load_async_to_lds_b128 %0, %1, %2"
                 :: "v"(sx1b + (unsigned)i*16u), "v"((unsigned)i*16u), "s"(xa)
                 : "memory");
  for (int i = t; i < WW/4; i += 256)
    asm volatile("global_load_async_to_lds_b128 %0, %1, %2"
                 :: "v"(swb + (unsigned)i*16u), "v"((unsigned)i*16u), "s"(wrow)
                 : "memory");
  if (t < 16) s_xb[t] = x2[(size_t)e * X2W + t];
  asm volatile("s_wait_asynccnt 0" ::: "memory");
  __syncthreads();

  // --- build B_e[i,col] = sum_j x2[e,j] * W3J[k][i][j][kk] -----------------
  if (t < 243) {
    int g, i, nidx;
    if (t < 16)      { g = 0; i = 0;          nidx = t; }
    else if (t < 88) { int r = t - 16; g = 1; i = r / 24; nidx = r % 24; }
    else             { int r = t - 88; g = 2; i = r / 31; nidx = r % 31; }
    int k  = COLK[g][nidx];
    int kk = COLKK[g][nidx];
    const float* w3 = g_w3j + k * W3JSTR + i * 49 + kk;
    int joff = K_I2OFF[k], nj = K_NJ[k];
    float acc = 0.f;
    for (int j = 0; j < nj; ++j) acc += s_xb[joff + j] * w3[j * 7];
    s_B[t] = acc;                // layout == GRP_BOFF[g] + i*ncol + nidx
  }
  __syncthreads();

  const int lane = t & 31, wave = t >> 5;
  const int half = lane >> 4, l16 = lane & 15;
  const int u0 = wave * 16;

  const int GRP_NI[3]   = {1, 3, 5};
  const int GRP_OFF1[3] = {0, 128, 512};
  const int GRP_NCOL[3] = {16, 24, 31};
  const int GRP_BOFF[3] = {0, 16, 88};
  const int GRP_NT[3]   = {1, 2, 2};
  const int GRP_NC[3]   = {1, 1, 2};

#pragma unroll
  for (int g = 0; g < 3; ++g) {
    const int nI = GRP_NI[g], off1 = GRP_OFF1[g];
    const int ncol = GRP_NCOL[g], boff = GRP_BOFF[g];
    const int urow = u0 + l16;

    // A operand (ISA 16x4 f32 layout: v0 = K{0,2}, v1 = K{1,3}); branchless OOB.
    v2f av[2];
#pragma unroll
    for (int c = 0; c < 2; ++c) {
      if (c < GRP_NC[g]) {
        int i0 = 4*c + 2*half;
        int ia = (i0     < nI) ? i0     : 0;
        int ib = (i0 + 1 < nI) ? i0 + 1 : 0;
        float a0 = s_x1[off1 + urow * nI + ia];
        float a1 = s_x1[off1 + urow * nI + ib];
        av[c][0] = (i0     < nI) ? a0 : 0.f;
        av[c][1] = (i0 + 1 < nI) ? a1 : 0.f;
      } else { av[c][0] = 0.f; av[c][1] = 0.f; }
    }

#pragma unroll
    for (int nt = 0; nt < 2; ++nt) {
      if (nt < GRP_NT[g]) {
        const int col   = nt * 16 + l16;
        const bool cok  = col < ncol;
        const int colc  = cok ? col : 0;

        v8f acc = {0.f,0.f,0.f,0.f,0.f,0.f,0.f,0.f};
#pragma unroll
        for (int c = 0; c < 2; ++c) {
          if (c < GRP_NC[g]) {
            int r0 = 4*c + 2*half;
            int rb0 = (r0     < nI) ? r0     : 0;
            int rb1 = (r0 + 1 < nI) ? r0 + 1 : 0;
            float b0 = s_B[boff + rb0 * ncol + colc];
            float b1 = s_B[boff + rb1 * ncol + colc];
            v2f bv;
            bv[0] = (cok && r0     < nI) ? b0 : 0.f;
            bv[1] = (cok && r0 + 1 < nI) ? b1 : 0.f;
            acc = __builtin_amdgcn_wmma_f32_16x16x4_f32(
                false, av[c], false, bv, (short)0, acc, false, false);
          }
        }

        // Epilogue: D[m,col] -> s_out[K_OUT + u*nk + kk], scaled by weight.
        const int k  = COLK[g][colc];
        const int kk = COLKK[g][colc];
        const int nk = K_NK[k], outo = K_OUT[k];
#pragma unroll
        for (int r = 0; r < 8; ++r) {
          int u = u0 + r + 8 * half;
          float wv = s_w[k * MULW + u];
          if (cok) s_out[outo + u * nk + kk] = wv * acc[r];
        }
      }
    }
  }

  __syncthreads();

  // --- coalesced async write-out: LDS -> HBM in B128 chunks ----------------
  for (int i = t; i < OUTW/4; i += 256)
    asm volatile("global_store_async_from_lds_b128 %0, %1, %2"
                 :: "v"((unsigned)i*16u), "v"(sob + (unsigned)i*16u), "s"(orow)
                 : "memory");
  asm volatile("s_wait_asynccnt 0" ::: "memory");
}

// ---------------------------------------------------------------------------
extern "C" void kernel_launch(void* const* d_in, const int* in_sizes, int n_in,
                              void* d_out, int out_size, void* d_ws, size_t ws_size,
                              hipStream_t stream) {
  const float* x1 = (const float*)d_in[0];
  const float* x2 = (const float*)d_in[1];
  const float* w  = (const float*)d_in[2];
  float* out = (float*)d_out;
  int n = in_sizes[0] / X1W;    // E

  w3j_setup_kernel<<<1, 32, 0, stream>>>();
  tp_wmma_kernel<<<n, 256, 0, stream>>>(x1, x2, w, out, n);
}